// GraphVQ_49417893707906
// MI455X (gfx1250) — compile-verified
//
#include <hip/hip_runtime.h>
#include <math.h>

typedef __attribute__((ext_vector_type(2))) float v2f;
typedef __attribute__((ext_vector_type(8))) float v8f;

#define B_TOT 524288
#define DHALF 64
#define KDIM 128
#define NCODE 64
#define ROWS_PER_BLOCK 128
#define NBLOCKS (B_TOT / ROWS_PER_BLOCK) /* 4096 */
#define ZOUT_FLOATS ((size_t)B_TOT * KDIM) /* 67,108,864 floats (complex interleaved) */
#define LOSS_OFF ZOUT_FLOATS
#define IDX_OFF (ZOUT_FLOATS + 1)

__device__ __forceinline__ float shfl_f(float v, int srcLane) {
  return __builtin_bit_cast(float,
      __builtin_amdgcn_ds_bpermute(srcLane << 2, __builtin_bit_cast(int, v)));
}
__device__ __forceinline__ int shfl_i(int v, int srcLane) {
  return __builtin_amdgcn_ds_bpermute(srcLane << 2, v);
}

__global__ __launch_bounds__(256) void graphvq_main(
    const float* __restrict__ z_real, const float* __restrict__ z_imag,
    const float* __restrict__ codebook, const float* __restrict__ adj,
    const int* __restrict__ prev, float* __restrict__ outF,
    float* __restrict__ partial)
{
  // B fragments of C^T in WMMA layout: slot (kk, nt, lane) -> float2 {C[n][k], C[n][k+1]}
  // with n = nt*16 + (lane&15), k = kk*4 + 2*(lane>>4)
  __shared__ v2f   cbfrag[4096];     // 32 KB
  __shared__ float sig[64 * 65];     // padded sigmoid(adj) table, 16.6 KB
  __shared__ float cnorm[NCODE];
  __shared__ int   idx_lds[ROWS_PER_BLOCK];
  __shared__ float wpart[8];

  const int tid = threadIdx.x;

  // ---- stage codebook into B-fragment layout ----
  for (int s = tid; s < 4096; s += 256) {
    const int l = s & 31, t = s >> 5;
    const int kk = t >> 2, nt = t & 3;
    const int n = nt * 16 + (l & 15);
    const int k = kk * 4 + 2 * (l >> 4);
    cbfrag[s] = *(const v2f*)(codebook + n * KDIM + k);
  }
  // ---- stage sigmoid(adj) ----
  for (int s = tid; s < 4096; s += 256) {
    const int r = s >> 6, c = s & 63;
    sig[r * 65 + c] = 1.0f / (1.0f + __expf(-adj[s]));
  }
  // ---- codebook squared norms ----
  if (tid < NCODE) {
    const float* cr = codebook + tid * KDIM;
    float sum = 0.f;
#pragma unroll 4
    for (int k = 0; k < KDIM; ++k) { const float v = cr[k]; sum += v * v; }
    cnorm[tid] = sum;
  }
  __syncthreads();

  const int w    = tid >> 5;      // wave id in block (0..7)
  const int lane = tid & 31;
  const int g    = lane >> 4;     // half-wave group
  const int rl   = lane & 15;     // local row for A fragments
  const int rowBase = blockIdx.x * ROWS_PER_BLOCK + w * 16;
  const size_t rowOfs = (size_t)(rowBase + rl) * DHALF;

  // ---- load A fragments (16x4 f32 layout), keep in registers for reuse ----
  v2f a[32];
#pragma unroll
  for (int kk = 0; kk < 16; ++kk)
    a[kk] = *(const v2f*)(z_real + rowOfs + kk * 4 + 2 * g);
#pragma unroll
  for (int kk = 0; kk < 16; ++kk)
    a[16 + kk] = *(const v2f*)(z_imag + rowOfs + kk * 4 + 2 * g);

  // ---- zf @ C^T via V_WMMA_F32_16X16X4_F32: 32 k-steps x 4 n-tiles ----
  v8f acc0 = {}, acc1 = {}, acc2 = {}, acc3 = {};
#pragma unroll
  for (int kk = 0; kk < 32; ++kk) {
    const v2f* bb = cbfrag + kk * 128 + lane;
    acc0 = __builtin_amdgcn_wmma_f32_16x16x4_f32(false, a[kk], false, bb[0],
                                                 (short)0, acc0, false, false);
    acc1 = __builtin_amdgcn_wmma_f32_16x16x4_f32(false, a[kk], false, bb[32],
                                                 (short)0, acc1, false, false);
    acc2 = __builtin_amdgcn_wmma_f32_16x16x4_f32(false, a[kk], false, bb[64],
                                                 (short)0, acc2, false, false);
    acc3 = __builtin_amdgcn_wmma_f32_16x16x4_f32(false, a[kk], false, bb[96],
                                                 (short)0, acc3, false, false);
  }

  // ---- argmin epilogue (||zf||^2 dropped: constant per row) ----
  const int prevv = prev[rowBase + rl];   // lane l holds prev of row (l&15)

#pragma unroll
  for (int j = 0; j < 8; ++j) {
    const int m = j + 8 * g;              // C/D layout: VGPR j, half g -> row m
    const int p = shfl_i(prevv, m);       // prev index for this row
    float bestV = 3.4e38f;
    int   bestN = 0;
    const float av[4] = { acc0[j], acc1[j], acc2[j], acc3[j] };
#pragma unroll
    for (int nt = 0; nt < 4; ++nt) {
      const int n = nt * 16 + rl;
      const float v = cnorm[n] - 2.f * av[nt] - 0.5f * sig[p * 65 + n];
      if (v < bestV) { bestV = v; bestN = n; }   // ascending n keeps first min
    }
    // reduce over the 16 lanes of this half (first-occurrence tie-break)
#pragma unroll
    for (int off = 1; off <= 8; off <<= 1) {
      const float ov = shfl_f(bestV, lane ^ off);
      const int   on = shfl_i(bestN, lane ^ off);
      if (ov < bestV || (ov == bestV && on < bestN)) { bestV = ov; bestN = on; }
    }
    if (rl == j) {                         // lanes j and j+16 own rows j, j+8
      idx_lds[w * 16 + m] = bestN;
      outF[IDX_OFF + (size_t)(rowBase + m)] = (float)bestN;
    }
  }
  __syncthreads();

  // ---- loss: squared error from A fragments (zf read only once from HBM) ----
  float err = 0.f;
  {
    const int idxr = idx_lds[w * 16 + rl];           // own row's code index
    const int slotN = (idxr >> 4) * 32 + (idxr & 15) + 16 * g;
#pragma unroll
    for (int kk = 0; kk < 32; ++kk) {
      const v2f c2 = cbfrag[kk * 128 + slotN];
      const float dx = c2.x - a[kk].x;
      const float dy = c2.y - a[kk].y;
      err += dx * dx + dy * dy;
    }
  }
#pragma unroll
  for (int off = 1; off < 32; off <<= 1) err += shfl_f(err, lane ^ off);
  if (lane == 0) wpart[w] = err;

  // ---- z_out: interleaved complex (re, im) from gathered codebook rows ----
#pragma unroll
  for (int m = 0; m < 16; ++m) {
    const int idxm = idx_lds[w * 16 + m];            // uniform within wave
    const float* cr = codebook + idxm * KDIM;        // L2-resident (32 KB)
    const size_t obase = (size_t)(rowBase + m) * KDIM;
#pragma unroll
    for (int dd = 0; dd < 2; ++dd) {
      const int d = lane + 32 * dd;
      v2f o = { cr[d], cr[DHALF + d] };
      *(v2f*)(outF + obase + 2 * d) = o;             // coalesced 8B stores
    }
  }

  __syncthreads();
  if (tid == 0) {
    float s = 0.f;
#pragma unroll
    for (int i = 0; i < 8; ++i) s += wpart[i];
    partial[blockIdx.x] = s;
  }
}

__global__ __launch_bounds__(256) void graphvq_loss(
    const float* __restrict__ partial, float* __restrict__ outF)
{
  __shared__ float red[256];
  float s = 0.f;
  for (int i = threadIdx.x; i < NBLOCKS; i += 256) s += partial[i];
  red[threadIdx.x] = s;
  __syncthreads();
  for (int step = 128; step > 0; step >>= 1) {
    if (threadIdx.x < step) red[threadIdx.x] += red[threadIdx.x + step];
    __syncthreads();
  }
  if (threadIdx.x == 0)
    outF[LOSS_OFF] = 1.25f * red[0] / ((float)B_TOT * (float)KDIM);
}

extern "C" void kernel_launch(void* const* d_in, const int* in_sizes, int n_in,
                              void* d_out, int out_size, void* d_ws, size_t ws_size,
                              hipStream_t stream) {
  const float* z_real   = (const float*)d_in[0];
  const float* z_imag   = (const float*)d_in[1];
  const float* codebook = (const float*)d_in[2];
  const float* adj      = (const float*)d_in[3];
  const int*   prev     = (const int*)d_in[4];
  float* outF    = (float*)d_out;
  float* partial = (float*)d_ws;   // NBLOCKS floats of scratch

  graphvq_main<<<NBLOCKS, 256, 0, stream>>>(z_real, z_imag, codebook, adj,
                                            prev, outF, partial);
  graphvq_loss<<<1, 256, 0, stream>>>(partial, outF);
}